// MultiHeadClassifier_22832046146100
// MI455X (gfx1250) — compile-verified
//
#include <hip/hip_runtime.h>

typedef __attribute__((ext_vector_type(2))) float v2f;
typedef __attribute__((ext_vector_type(4))) float v4f;
typedef __attribute__((ext_vector_type(8))) float v8f;

#define PAD_SENTINEL 0x80000000

// ---------------- binning prologue ----------------

__global__ void mh_init(int* __restrict__ counts, int* __restrict__ perm, int permN) {
  int i = blockIdx.x * blockDim.x + threadIdx.x;
  int stride = gridDim.x * blockDim.x;
  for (int j = i; j < permN; j += stride) perm[j] = (int)PAD_SENTINEL;
  if (i < 8) counts[i] = 0;
}

__global__ void mh_count(const int* __restrict__ labels, int n, int* __restrict__ counts) {
  int i = blockIdx.x * blockDim.x + threadIdx.x;
  int stride = gridDim.x * blockDim.x;
  for (int j = i; j < n; j += stride) atomicAdd(&counts[labels[j]], 1);
}

__global__ void mh_scan(const int* __restrict__ counts, int* __restrict__ base,
                        int* __restrict__ fill) {
  if (threadIdx.x == 0 && blockIdx.x == 0) {
    int acc = 0;
    for (int t = 0; t < 8; ++t) {
      base[t] = acc;
      fill[t] = acc;
      acc += (counts[t] + 63) & ~63;   // pad each bucket to a 64-row tile
    }
    base[8] = acc;
  }
}

__global__ void mh_scatter(const int* __restrict__ labels, int n,
                           int* __restrict__ fill, int* __restrict__ perm) {
  int i = blockIdx.x * blockDim.x + threadIdx.x;
  int stride = gridDim.x * blockDim.x;
  for (int j = i; j < n; j += stride) {
    int r = atomicAdd(&fill[labels[j]], 1);
    perm[r] = j;
  }
}

// ------- grouped GEMM: 64 rows x 1024 x 32 classes per block, f32 WMMA -------

__global__ __launch_bounds__(128)
void mh_gemm(const float* __restrict__ x, const float* __restrict__ W,
             const float* __restrict__ bias, const int* __restrict__ base,
             const int* __restrict__ perm, float* __restrict__ out)
{
  constexpr int D  = 1024;
  constexpr int C  = 32;
  constexpr int KC = 256;        // K-chunk staged in LDS
  constexpr int P  = KC + 4;     // pitch ≡ 4 (mod 64) dwords -> conflict-free b64 frags
  constexpr int TM = 64;         // rows per block

  __shared__ float xs [TM * P];  // gathered x tile   (~65 KB)
  __shared__ float wsh[C  * P];  // W_t chunk         (~32.5 KB)
  __shared__ int   rowIdx[TM];

  int tileBase = blockIdx.x * TM;
  int t = -1;
#pragma unroll
  for (int q = 0; q < 8; ++q)
    if (tileBase >= base[q] && tileBase < base[q + 1]) t = q;
  if (t < 0) return;             // uniform per block, before any barrier

  const float* Wt = W + (size_t)t * C * D;

  const int tid   = threadIdx.x;
  const int lane  = tid & 31;
  const int wave  = tid >> 5;    // 0..3 -> which 16-row tile
  const int half  = lane >> 4;   // K sub-half per ISA f32 A/B layout
  const int r     = lane & 15;
  const int rbase = tid >> 6;    // staging: row parity (0/1)
  const int c4    = tid & 63;    // staging: float4 column

  if (tid < TM) rowIdx[tid] = perm[tileBase + tid];
  __syncthreads();

  // Per-thread x-row byte offsets for staging. Padding slots (bit31 set) are
  // masked to row 0: their garbage only affects their own (never stored) rows.
  unsigned xoff[TM / 2];
#pragma unroll
  for (int s = 0; s < TM / 2; ++s) {
    int g = rowIdx[rbase + 2 * s] & 0x7fffffff;
    xoff[s] = (unsigned)g * (unsigned)(D * 4) + (unsigned)c4 * 16u;
  }

  v8f acc0 = {0.f, 0.f, 0.f, 0.f, 0.f, 0.f, 0.f, 0.f};
  v8f acc1 = {0.f, 0.f, 0.f, 0.f, 0.f, 0.f, 0.f, 0.f};

  for (int k0 = 0; k0 < D; k0 += KC) {
    // ---- stage x tile: 64 rows x 64 float4, batches of 8 loads in flight ----
    const char* xb = (const char*)x + (size_t)k0 * 4;
#pragma unroll
    for (int gq = 0; gq < 4; ++gq) {
      v4f tv[8];
#pragma unroll
      for (int s = 0; s < 8; ++s)
        tv[s] = __builtin_nontemporal_load(
            (const v4f*)(xb + xoff[gq * 8 + s]));
#pragma unroll
      for (int s = 0; s < 8; ++s)
        *(v4f*)&xs[(rbase + 2 * (gq * 8 + s)) * P + c4 * 4] = tv[s];
    }
    // ---- stage W_t chunk: 32 classes x 64 float4 (L2-resident, cached) ----
#pragma unroll
    for (int gq = 0; gq < 2; ++gq) {
      v4f tv[8];
#pragma unroll
      for (int s = 0; s < 8; ++s)
        tv[s] = *(const v4f*)(Wt + (size_t)(rbase + 2 * (gq * 8 + s)) * D +
                              k0 + c4 * 4);
#pragma unroll
      for (int s = 0; s < 8; ++s)
        *(v4f*)&wsh[(rbase + 2 * (gq * 8 + s)) * P + c4 * 4] = tv[s];
    }
    __syncthreads();

    // A frag (16x4 f32): lane<16 -> (row r, k,k+1); lane>=16 -> (row r, k+2,k+3)
    const float* ax = &xs [(wave * 16 + r) * P + 2 * half];
    // B frags for both 16-class tiles (shared A -> 3 frag loads per 2 WMMAs)
    const float* b0 = &wsh[(r)      * P + 2 * half];
    const float* b1 = &wsh[(16 + r) * P + 2 * half];

#pragma unroll 8
    for (int kk = 0; kk < KC; kk += 4) {
      v2f a   = *(const v2f*)(ax + kk);
      v2f bb0 = *(const v2f*)(b0 + kk);
      v2f bb1 = *(const v2f*)(b1 + kk);
      acc0 = __builtin_amdgcn_wmma_f32_16x16x4_f32(
          false, a, false, bb0, (short)0, acc0, false, false);
      acc1 = __builtin_amdgcn_wmma_f32_16x16x4_f32(
          false, a, false, bb1, (short)0, acc1, false, false);
    }
    __syncthreads();
  }

  // epilogue: D layout VGPR j -> M=j (lanes 0-15) / M=j+8 (lanes 16-31), N=lane&15
  float bv0 = bias[t * C + r];
  float bv1 = bias[t * C + 16 + r];
#pragma unroll
  for (int j = 0; j < 8; ++j) {
    int m = wave * 16 + j + 8 * half;
    int g = rowIdx[m];
    if (g >= 0) {
      __builtin_nontemporal_store(acc0[j] + bv0, &out[(size_t)g * C + r]);
      __builtin_nontemporal_store(acc1[j] + bv1, &out[(size_t)g * C + 16 + r]);
    }
  }
}

// ---------------- fallback (shape mismatch / tiny workspace) ----------------

__global__ void mh_naive(const float* __restrict__ x, const int* __restrict__ labels,
                         const float* __restrict__ W, const float* __restrict__ bias,
                         float* __restrict__ out, int B, int D, int C) {
  int i = blockIdx.x * blockDim.x + threadIdx.x;
  if (i >= B * C) return;
  int b = i / C, c = i % C;
  int t = labels[b];
  const float* w  = W + ((size_t)t * C + c) * D;
  const float* xr = x + (size_t)b * D;
  float s = 0.f;
  for (int d = 0; d < D; ++d) s = fmaf(xr[d], w[d], s);
  out[i] = s + bias[t * C + c];
}

// ---------------- host ----------------

extern "C" void kernel_launch(void* const* d_in, const int* in_sizes, int n_in,
                              void* d_out, int out_size, void* d_ws, size_t ws_size,
                              hipStream_t stream) {
  const float* x      = (const float*)d_in[0];
  const int*   labels = (const int*)  d_in[1];
  const float* W      = (const float*)d_in[2];
  const float* bias   = (const float*)d_in[3];
  float*       out    = (float*)d_out;

  int B = in_sizes[1];
  int D = (B > 0) ? in_sizes[0] / B : 0;
  int C = (B > 0) ? out_size / B : 0;
  int T = (C > 0) ? in_sizes[3] / C : 0;

  size_t need = (size_t)(8 + 9 + 8 + B + 8 * 64) * sizeof(int);

  if (D == 1024 && C == 32 && T == 8 && ws_size >= need) {
    int* counts = (int*)d_ws;      // 8
    int* base   = counts + 8;      // 9
    int* fill   = base + 9;        // 8
    int* perm   = fill + 8;        // B + 512
    int permN   = B + 8 * 64;

    mh_init   <<<1024, 256, 0, stream>>>(counts, perm, permN);
    mh_count  <<<1024, 256, 0, stream>>>(labels, B, counts);
    mh_scan   <<<1,    32,  0, stream>>>(counts, base, fill);
    mh_scatter<<<1024, 256, 0, stream>>>(labels, B, fill, perm);

    int tiles = permN / 64;
    mh_gemm<<<tiles, 128, 0, stream>>>(x, W, bias, base, perm, out);
  } else {
    int total = B * C;
    mh_naive<<<(total + 255) / 256, 256, 0, stream>>>(x, labels, W, bias, out, B, D, C);
  }
}